// HeteroGNN_54193897341585
// MI455X (gfx1250) — compile-verified
//
#include <hip/hip_runtime.h>
#include <math.h>

#define HDIM 128
#define SLOPE 0.2f

typedef __attribute__((ext_vector_type(2))) float v2f;
typedef __attribute__((ext_vector_type(8))) float v8f;

// ---------------------------------------------------------------------------
// Dense GEMM  Y[N,128] = X[N,128] @ W[128,128]  via V_WMMA_F32_16X16X4_F32.
// One wave per 16x16 output tile; K loop of 32 WMMAs.
// A(16x4 f32) layout: lane 0-15 M=0..15; VGPR0 = K=kOff, VGPR1 = K=kOff+1,
// kOff = 2*(lane>>4).  B(4x16): row striped across lanes within a VGPR.
// C/D: VGPR j -> (M = j + 8*(lane>>4), N = lane&15).
// N is a multiple of 16 (50000 = 3125*16) so EXEC is all-ones for WMMA.
// ---------------------------------------------------------------------------
__global__ __launch_bounds__(256) void wmma_gemm_nk(
    const float* __restrict__ X, const float* __restrict__ Wm,
    float* __restrict__ Y, int nTilesM) {
  int w = blockIdx.x * (blockDim.x >> 5) + (threadIdx.x >> 5);
  int tileM = w >> 3;            // H/16 = 8 N-tiles per M strip
  int tileN = w & 7;
  if (tileM >= nTilesM) return;  // wave-uniform
  int lane = threadIdx.x & 31;
  int idx  = lane & 15;
  int kOff = (lane >> 4) << 1;   // 0 or 2
  const float* Arow = X  + (size_t)(tileM * 16 + idx) * HDIM;
  const float* Bcol = Wm + tileN * 16 + idx;
  v8f acc = {};
#pragma unroll
  for (int k = 0; k < HDIM; k += 4) {
    v2f a = *(const v2f*)(Arow + k + kOff);     // 8B contiguous load
    v2f b;
    b.x = Bcol[(size_t)(k + kOff)     * HDIM];
    b.y = Bcol[(size_t)(k + kOff + 1) * HDIM];
    acc = __builtin_amdgcn_wmma_f32_16x16x4_f32(
        false, a, false, b, (short)0, acc, false, false);
  }
  int row = tileM * 16 + (lane >> 4) * 8;
  int col = tileN * 16 + idx;
#pragma unroll
  for (int j = 0; j < 8; ++j)
    Y[(size_t)(row + j) * HDIM + col] = acc[j];
}

// out[n] = dot(Hm[n,:], a[:])  -- one wave per node, float4 loads + shfl tree
__global__ __launch_bounds__(256) void rowdot(
    const float* __restrict__ Hm, const float* __restrict__ a,
    float* __restrict__ out, int N) {
  int n = blockIdx.x * (blockDim.x >> 5) + (threadIdx.x >> 5);
  if (n >= N) return;
  int lane = threadIdx.x & 31;
  float4 h4 = *(const float4*)(Hm + (size_t)n * HDIM + lane * 4);
  float4 a4 = *(const float4*)(a + lane * 4);
  float v = h4.x * a4.x + h4.y * a4.y + h4.z * a4.z + h4.w * a4.w;
#pragma unroll
  for (int off = 16; off; off >>= 1) v += __shfl_xor(v, off, 32);
  if (lane == 0) out[n] = v;
}

__global__ void zero_f(float* __restrict__ p, long long n) {
  long long i = (long long)blockIdx.x * blockDim.x + threadIdx.x;
  if (i < n) p[i] = 0.f;
}

// order-preserving float<->u32 key (for atomic segment-max)
__device__ __forceinline__ unsigned fenc(float x) {
  unsigned b = __float_as_uint(x);
  return (b & 0x80000000u) ? ~b : (b | 0x80000000u);
}
__device__ __forceinline__ float fdec(unsigned k) {
  return __uint_as_float((k & 0x80000000u) ? (k & 0x7FFFFFFFu) : ~k);
}

__global__ void init_ms(unsigned* __restrict__ mkey, float* __restrict__ msum, int N) {
  int n = blockIdx.x * blockDim.x + threadIdx.x;
  if (n < N) { mkey[n] = 0x007FFFFFu; /* enc(-inf) */ msum[n] = 0.f; }
}

__global__ void edge_score(const int* __restrict__ src, const int* __restrict__ dst,
                           const float* __restrict__ as_, const float* __restrict__ ad_,
                           float* __restrict__ e, unsigned* __restrict__ mkey, int E) {
  int i = blockIdx.x * blockDim.x + threadIdx.x;
  if (i >= E) return;
  float v = as_[src[i]] + ad_[dst[i]];
  v = v > 0.f ? v : SLOPE * v;           // leaky_relu
  e[i] = v;
  atomicMax(&mkey[dst[i]], fenc(v));
}

__global__ void fix_max(const unsigned* __restrict__ mkey, float* __restrict__ mval, int N) {
  int n = blockIdx.x * blockDim.x + threadIdx.x;
  if (n >= N) return;
  float v = fdec(mkey[n]);
  mval[n] = isfinite(v) ? v : 0.f;       // empty segments -> 0
}

__global__ void edge_exp(const int* __restrict__ dst, const float* __restrict__ mval,
                         float* __restrict__ e, float* __restrict__ msum, int E) {
  int i = blockIdx.x * blockDim.x + threadIdx.x;
  if (i >= E) return;
  float p = __expf(e[i] - mval[dst[i]]);
  e[i] = p;                              // overwrite score with exp in place
  atomicAdd(&msum[dst[i]], p);
}

// one wave per edge: gather hS row (float4/lane), scatter-add alpha*row
__global__ __launch_bounds__(256) void edge_scatter(
    const int* __restrict__ src, const int* __restrict__ dst,
    const float* __restrict__ p, const float* __restrict__ msum,
    const float* __restrict__ hS, float* __restrict__ out, int E) {
  int i = blockIdx.x * (blockDim.x >> 5) + (threadIdx.x >> 5);
  if (i >= E) return;
  int lane = threadIdx.x & 31;
  int s = src[i], d = dst[i];
  float alpha = p[i] / (msum[d] + 1e-16f);
  float4 h4 = *(const float4*)(hS + (size_t)s * HDIM + lane * 4);
  float* o = out + (size_t)d * HDIM + lane * 4;
  atomicAdd(o + 0, alpha * h4.x);
  atomicAdd(o + 1, alpha * h4.y);
  atomicAdd(o + 2, alpha * h4.z);
  atomicAdd(o + 3, alpha * h4.w);
}

// out = relu(acc + b1 + (b2?b2:0)), elementwise over N*H (in place allowed)
__global__ void combine_relu(const float* __restrict__ acc,
                             const float* __restrict__ b1, const float* __restrict__ b2,
                             float* __restrict__ out, int N) {
  long long t = (long long)blockIdx.x * blockDim.x + threadIdx.x;
  if (t >= (long long)N * HDIM) return;
  int h = (int)(t & (HDIM - 1));
  float v = acc[t] + b1[h] + (b2 ? b2[h] : 0.f);
  out[t] = v > 0.f ? v : 0.f;
}

extern "C" void kernel_launch(void* const* d_in, const int* in_sizes, int n_in,
                              void* d_out, int out_size, void* d_ws, size_t ws_size,
                              hipStream_t stream) {
  const float* xg0   = (const float*)d_in[0];
  const float* xp0   = (const float*)d_in[1];
  const int*   egp   = (const int*)d_in[2];
  const int*   epg   = (const int*)d_in[3];
  const int*   egg   = (const int*)d_in[4];
  const float* Wall  = (const float*)d_in[5];
  const float* att_s = (const float*)d_in[6];
  const float* att_d = (const float*)d_in[7];
  const float* bias  = (const float*)d_in[8];

  const int NG = in_sizes[0] / HDIM;
  const int NP = in_sizes[1] / HDIM;
  const int E  = in_sizes[2] / 2;
  const int R = 3, L = 2;
  const size_t nmax = (size_t)(NG > NP ? NG : NP);

  // workspace layout (floats)
  float* ws = (float*)d_ws;
  float*    hA   = ws;            ws += nmax * HDIM;   // src projection
  float*    hB   = ws;            ws += nmax * HDIM;   // dst projection
  float*    x1g  = ws;            ws += (size_t)NG * HDIM; // layer-1 gene acts
  float*    x1p  = ws;            ws += (size_t)NP * HDIM; // layer-1 prot acts
  float*    as_  = ws;            ws += nmax;
  float*    ad_  = ws;            ws += nmax;
  unsigned* mkey = (unsigned*)ws; ws += nmax;
  float*    mval = ws;            ws += nmax;
  float*    msum = ws;            ws += nmax;
  float*    ebuf = ws;            ws += (size_t)E;

  float* outg = (float*)d_out;                    // final gene  [NG*H]
  float* outp = (float*)d_out + (size_t)NG * HDIM;// final prot  [NP*H]

  const int TB = 256;
  auto gcd = [](long long n) { return (unsigned)((n + TB - 1) / TB); };

  for (int l = 0; l < L; ++l) {
    const float* xg = (l == 0) ? xg0 : x1g;
    const float* xp = (l == 0) ? xp0 : x1p;
    float* accg = (l == L - 1) ? outg : x1g;
    float* accp = (l == L - 1) ? outp : x1p;

    zero_f<<<gcd((long long)NG * HDIM), TB, 0, stream>>>(accg, (long long)NG * HDIM);
    zero_f<<<gcd((long long)NP * HDIM), TB, 0, stream>>>(accp, (long long)NP * HDIM);

    auto do_rel = [&](int r, const float* xs, int Ns, const float* xd, int Nd,
                      const int* edge, float* acc, bool same) {
      const float* Wr  = Wall  + (size_t)(l * R + r) * HDIM * HDIM;
      const float* asr = att_s + (size_t)(l * R + r) * HDIM;
      const float* adr = att_d + (size_t)(l * R + r) * HDIM;
      // projections (WMMA GEMMs); N is a multiple of 16
      wmma_gemm_nk<<<Ns / 16, TB, 0, stream>>>(xs, Wr, hA, Ns / 16);
      const float* hS = hA;
      const float* hD = hA;
      if (!same) {
        wmma_gemm_nk<<<Nd / 16, TB, 0, stream>>>(xd, Wr, hB, Nd / 16);
        hD = hB;
      }
      // per-node attention scalars
      rowdot<<<(Ns + 7) / 8, TB, 0, stream>>>(hS, asr, as_, Ns);
      rowdot<<<(Nd + 7) / 8, TB, 0, stream>>>(hD, adr, ad_, Nd);
      // segment softmax
      init_ms<<<gcd(Nd), TB, 0, stream>>>(mkey, msum, Nd);
      edge_score<<<gcd(E), TB, 0, stream>>>(edge, edge + E, as_, ad_, ebuf, mkey, E);
      fix_max<<<gcd(Nd), TB, 0, stream>>>(mkey, mval, Nd);
      edge_exp<<<gcd(E), TB, 0, stream>>>(edge + E, mval, ebuf, msum, E);
      // weighted scatter-add of source rows
      edge_scatter<<<(E + 7) / 8, TB, 0, stream>>>(edge, edge + E, ebuf, msum, hS, acc, E);
    };

    do_rel(0, xg, NG, xp, NP, egp, accp, false);  // gene -> protein
    do_rel(1, xp, NP, xg, NG, epg, accg, false);  // protein -> gene
    do_rel(2, xg, NG, xg, NG, egg, accg, true);   // gene -> gene

    const float* b0 = bias + (size_t)(l * R + 0) * HDIM;
    const float* b1 = bias + (size_t)(l * R + 1) * HDIM;
    const float* b2 = bias + (size_t)(l * R + 2) * HDIM;
    combine_relu<<<gcd((long long)NG * HDIM), TB, 0, stream>>>(accg, b1, b2, accg, NG);
    combine_relu<<<gcd((long long)NP * HDIM), TB, 0, stream>>>(accp, b0, nullptr, accp, NP);
  }
}